// GNNModel_89704686944682
// MI455X (gfx1250) — compile-verified
//
#include <hip/hip_runtime.h>
#include <hip/hip_bf16.h>

typedef float v2f __attribute__((ext_vector_type(2)));
typedef float v8f __attribute__((ext_vector_type(8)));

#define DH 128

// ---------------- elementwise helpers ----------------

__global__ void __launch_bounds__(256) fill_kernel(float* __restrict__ p, float v, int n) {
  int i = blockIdx.x * 256 + threadIdx.x;
  if (i < n) p[i] = v;
}

__global__ void __launch_bounds__(256) edge_deg_kernel(const int* __restrict__ dst,
                                                       float* __restrict__ deg, int E) {
  int e = blockIdx.x * 256 + threadIdx.x;
  if (e < E) atomicAdd(&deg[dst[e]], 1.0f);
}

__global__ void __launch_bounds__(256) rsqrt_kernel(float* __restrict__ p, int n) {
  int i = blockIdx.x * 256 + threadIdx.x;
  if (i < n) p[i] = rsqrtf(p[i]);
}

// ---------------- fp32 WMMA GEMM: C[M x 128] = act(A0@W0 (+ A1@W1) + bias) ----------------
// Block = 256 threads = 8 waves; each wave computes a 16x128 output strip.
// W staged in LDS with k-interleave-by-4 layout: wlds[(k>>2)*512 + n*4 + (k&3)].
// -> B fragment {W[k][n], W[k+1][n]} is a contiguous 8B pair (single ds_load_b64,
//    no register shuffles) and lane halves hit disjoint LDS banks (conflict-free).

__global__ void __launch_bounds__(256)
gemm128_wmma(const float* __restrict__ A0, const float* __restrict__ W0,
             const float* __restrict__ A1, const float* __restrict__ W1,
             const float* __restrict__ bias, float* __restrict__ C,
             int M, int do_relu)
{
  __shared__ float wlds[DH * DH];
  const int lane = threadIdx.x & 31;
  const int wave = threadIdx.x >> 5;
  const int half = lane >> 4;   // 0: lanes 0-15, 1: lanes 16-31
  const int r    = lane & 15;
  const int row0 = blockIdx.x * 128 + wave * 16;

  v8f acc[8] = {};
  const int nphase = (A1 != nullptr) ? 2 : 1;

  for (int phase = 0; phase < nphase; ++phase) {
    const float* __restrict__ A = phase ? A1 : A0;
    const float* __restrict__ W = phase ? W1 : W0;

    __syncthreads();  // protect LDS reuse across phases
    // Stage W: lane gathers a 4-row column slice (coalesced 128B per row across
    // lanes), stores one contiguous b128 into the interleaved image.
    for (int t = threadIdx.x; t < 32 * DH; t += 256) {
      int kg = t >> 7;          // k-group: rows 4*kg .. 4*kg+3
      int n  = t & 127;
      const float* wp = W + (size_t)(4 * kg) * DH + n;
      float4 v = make_float4(wp[0], wp[DH], wp[2 * DH], wp[3 * DH]);
      *(float4*)&wlds[kg * 512 + n * 4] = v;
    }
    __syncthreads();

    int arow = row0 + r;
    if (arow >= M) arow = M - 1;  // clamp OOB loads; stores are guarded later
    const float* aptr = A + (size_t)arow * DH + 2 * half;

    #pragma unroll 2
    for (int k0 = 0; k0 < DH; k0 += 4) {
      // A 16x4 fragment: VGPR0/1 = A[r][k0+2*half], A[r][k0+2*half+1]
      v2f af = *(const v2f*)(aptr + k0);
      // B fragment base for this k-step and lane
      const float* bbase = &wlds[(k0 >> 2) * 512 + r * 4 + 2 * half];
      #pragma unroll
      for (int t = 0; t < 8; ++t) {
        // rows (k0+2*half, k0+2*half+1), col = t*16 + r : contiguous pair
        v2f bf = *(const v2f*)(bbase + t * 64);
        acc[t] = __builtin_amdgcn_wmma_f32_16x16x4_f32(
            false, af, false, bf, (short)0, acc[t], false, false);
      }
    }
  }

  // Epilogue: C/D layout -> row = row0 + v + 8*half, col = t*16 + r
  #pragma unroll
  for (int t = 0; t < 8; ++t) {
    const int n = t * 16 + r;
    const float bv = bias ? bias[n] : 0.0f;
    #pragma unroll
    for (int v = 0; v < 8; ++v) {
      const int row = row0 + v + 8 * half;
      if (row < M) {
        float val = acc[t][v] + bv;
        if (do_relu) val = fmaxf(val, 0.0f);
        C[(size_t)row * DH + n] = val;
      }
    }
  }
}

// ---------------- GCN propagation (symmetric normalized, self-loops folded in) ----------------
// One thread per (node/edge, 4-channel group): b128 feature loads.

__global__ void __launch_bounds__(256)
scatter_init_kernel(const float* __restrict__ m, const float* __restrict__ dinv,
                    const float* __restrict__ bias, float* __restrict__ out, int n)
{
  int idx = blockIdx.x * 256 + threadIdx.x;   // n*32 threads
  if (idx < n * 32) {
    int i = idx >> 5, c4 = idx & 31;
    float di = dinv[i];
    float w = di * di;
    float4 mv = *(const float4*)(m + ((size_t)idx << 2));
    float4 bv = *(const float4*)(bias + c4 * 4);
    float4 o;
    o.x = mv.x * w + bv.x;  o.y = mv.y * w + bv.y;
    o.z = mv.z * w + bv.z;  o.w = mv.w * w + bv.w;
    *(float4*)(out + ((size_t)idx << 2)) = o;
  }
}

__global__ void __launch_bounds__(256)
scatter_edge_kernel(const float* __restrict__ m, const int* __restrict__ src,
                    const int* __restrict__ dst, const float* __restrict__ dinv,
                    float* __restrict__ out, int E)
{
  int idx = blockIdx.x * 256 + threadIdx.x;   // E*32 threads
  if (idx < E * 32) {
    int e = idx >> 5, c4 = idx & 31;
    int s = src[e], d = dst[e];
    float w = dinv[s] * dinv[d];
    float4 mv = *(const float4*)(m + (size_t)s * DH + c4 * 4);
    float* op = out + (size_t)d * DH + c4 * 4;
    atomicAdd(op + 0, mv.x * w);
    atomicAdd(op + 1, mv.y * w);
    atomicAdd(op + 2, mv.z * w);
    atomicAdd(op + 3, mv.w * w);
  }
}

// ---------------- BatchNorm (training-mode batch stats) ----------------

__global__ void __launch_bounds__(256)
bn_stats_kernel(const float* __restrict__ m, float* __restrict__ sums, int n)
{
  int c4 = threadIdx.x & 31;   // channels 4*c4 .. 4*c4+3
  int rl = threadIdx.x >> 5;   // 0..7
  int rEnd = (blockIdx.x + 1) * 256;
  if (rEnd > n) rEnd = n;
  float s0 = 0.f, s1 = 0.f, s2 = 0.f, s3 = 0.f;
  float q0 = 0.f, q1 = 0.f, q2 = 0.f, q3 = 0.f;
  for (int row = blockIdx.x * 256 + rl; row < rEnd; row += 8) {
    float4 v = *(const float4*)(m + (size_t)row * DH + c4 * 4);
    s0 += v.x; s1 += v.y; s2 += v.z; s3 += v.w;
    q0 += v.x * v.x; q1 += v.y * v.y; q2 += v.z * v.z; q3 += v.w * v.w;
  }
  float* sp = sums + c4 * 4;
  atomicAdd(sp + 0, s0); atomicAdd(sp + 1, s1);
  atomicAdd(sp + 2, s2); atomicAdd(sp + 3, s3);
  sp += DH;
  atomicAdd(sp + 0, q0); atomicAdd(sp + 1, q1);
  atomicAdd(sp + 2, q2); atomicAdd(sp + 3, q3);
}

__global__ void __launch_bounds__(256)
bn_apply_kernel(const float* __restrict__ agg, const float* __restrict__ sums,
                const float* __restrict__ g, const float* __restrict__ b,
                float* __restrict__ h, int n, int add_res)
{
  int idx = blockIdx.x * 256 + threadIdx.x;   // n*32 threads
  if (idx < n * 32) {
    int c4 = idx & 31;
    float inv_n = 1.0f / (float)n;
    float4 sm = *(const float4*)(sums + c4 * 4);
    float4 sq = *(const float4*)(sums + DH + c4 * 4);
    float4 gv = *(const float4*)(g + c4 * 4);
    float4 bv = *(const float4*)(b + c4 * 4);
    float4 av = *(const float4*)(agg + ((size_t)idx << 2));
    float mu, var, istd, v;
    float4 o;
    mu = sm.x * inv_n; var = sq.x * inv_n - mu * mu; istd = rsqrtf(var + 1e-5f);
    v = (av.x - mu) * istd * gv.x + bv.x; o.x = fmaxf(v, 0.0f);
    mu = sm.y * inv_n; var = sq.y * inv_n - mu * mu; istd = rsqrtf(var + 1e-5f);
    v = (av.y - mu) * istd * gv.y + bv.y; o.y = fmaxf(v, 0.0f);
    mu = sm.z * inv_n; var = sq.z * inv_n - mu * mu; istd = rsqrtf(var + 1e-5f);
    v = (av.z - mu) * istd * gv.z + bv.z; o.z = fmaxf(v, 0.0f);
    mu = sm.w * inv_n; var = sq.w * inv_n - mu * mu; istd = rsqrtf(var + 1e-5f);
    v = (av.w - mu) * istd * gv.w + bv.w; o.w = fmaxf(v, 0.0f);
    float* hp = h + ((size_t)idx << 2);
    if (add_res) {
      float4 hv = *(const float4*)hp;
      o.x += hv.x; o.y += hv.y; o.z += hv.z; o.w += hv.w;
    }
    *(float4*)hp = o;
  }
}

// ---------------- graph pooling (segment mean + max) ----------------

__global__ void __launch_bounds__(256)
pool_scatter_kernel(const float* __restrict__ h, const int* __restrict__ batch,
                    float* __restrict__ gsum, float* __restrict__ gmax,
                    float* __restrict__ gcnt, int n)
{
  int idx = blockIdx.x * 256 + threadIdx.x;   // n*32 threads
  if (idx < n * 32) {
    int i = idx >> 5, c4 = idx & 31;
    int b = batch[i];
    float4 v = *(const float4*)(h + ((size_t)idx << 2));
    float* sp = gsum + (size_t)b * DH + c4 * 4;
    atomicAdd(sp + 0, v.x); atomicAdd(sp + 1, v.y);
    atomicAdd(sp + 2, v.z); atomicAdd(sp + 3, v.w);
    // post-ReLU values are >= 0 -> IEEE order == unsigned order, init 0 valid
    unsigned int* mp = (unsigned int*)(gmax + (size_t)b * DH + c4 * 4);
    atomicMax(mp + 0, __float_as_uint(v.x));
    atomicMax(mp + 1, __float_as_uint(v.y));
    atomicMax(mp + 2, __float_as_uint(v.z));
    atomicMax(mp + 3, __float_as_uint(v.w));
    if (c4 == 0) atomicAdd(&gcnt[b], 1.0f);
  }
}

__global__ void __launch_bounds__(256)
pool_final_kernel(const float* __restrict__ gsum, const float* __restrict__ gcnt,
                  float* __restrict__ gmean, int total)
{
  int idx = blockIdx.x * 256 + threadIdx.x;
  if (idx < total) gmean[idx] = gsum[idx] / gcnt[idx >> 7];
}

// ---------------- final tiny FC (512 x 10, K=128) ----------------

__global__ void __launch_bounds__(256)
fc2_kernel(const float* __restrict__ a, const float* __restrict__ w,
           const float* __restrict__ b, float* __restrict__ out, int G)
{
  int idx = blockIdx.x * 256 + threadIdx.x;
  if (idx < G * 10) {
    int g = idx / 10, o = idx - g * 10;
    float s = b[o];
    const float* ar = a + (size_t)g * DH;
    #pragma unroll 8
    for (int k = 0; k < DH; ++k) s += ar[k] * w[k * 10 + o];
    out[idx] = s;
  }
}

// ---------------- launcher ----------------

extern "C" void kernel_launch(void* const* d_in, const int* in_sizes, int n_in,
                              void* d_out, int out_size, void* d_ws, size_t ws_size,
                              hipStream_t stream) {
  (void)n_in; (void)out_size; (void)ws_size;
  const float* x      = (const float*)d_in[0];
  const int*   ei     = (const int*)d_in[1];
  const int*   batch  = (const int*)d_in[2];
  // d_in[3] = n_graphs (device scalar; value is fixed at 512 by the problem)
  const float* w_in   = (const float*)d_in[4];
  const float* b_in   = (const float*)d_in[5];
  const float* conv_w = (const float*)d_in[6];
  const float* conv_b = (const float*)d_in[7];
  const float* bn_g   = (const float*)d_in[8];
  const float* bn_b   = (const float*)d_in[9];
  const float* fc1_w  = (const float*)d_in[10];
  const float* fc1_b  = (const float*)d_in[11];
  const float* fc2_w  = (const float*)d_in[12];
  const float* fc2_b  = (const float*)d_in[13];

  const int N = in_sizes[0] / DH;   // 50000
  const int E = in_sizes[1] / 2;    // 600000
  const int G = 512;
  const int* src = ei;
  const int* dst = ei + E;

  size_t off = 0;
  auto carve = [&](size_t bytes) -> float* {
    float* p = (float*)((char*)d_ws + off);
    off += (bytes + 255) & ~(size_t)255;
    return p;
  };
  float* dinv  = carve((size_t)N * 4);            // deg then rsqrt in-place
  float* h     = carve((size_t)N * DH * 4);
  float* m     = carve((size_t)N * DH * 4);
  float* agg   = carve((size_t)N * DH * 4);
  float* sums  = carve((size_t)2 * DH * 4);
  float* gsum  = carve((size_t)G * DH * 4);
  float* gmaxb = carve((size_t)G * DH * 4);
  float* gcnt  = carve((size_t)G * 4);
  float* gmean = carve((size_t)G * DH * 4);
  float* fc1o  = carve((size_t)G * DH * 4);

  dim3 blk(256);
  auto cdiv = [](int a, int b) { return (a + b - 1) / b; };

  // degrees with self-loops -> dinv
  fill_kernel<<<cdiv(N, 256), blk, 0, stream>>>(dinv, 1.0f, N);
  edge_deg_kernel<<<cdiv(E, 256), blk, 0, stream>>>(dst, dinv, E);
  rsqrt_kernel<<<cdiv(N, 256), blk, 0, stream>>>(dinv, N);

  // h = relu(x @ w_in + b_in)
  gemm128_wmma<<<cdiv(N, 128), blk, 0, stream>>>(x, w_in, nullptr, nullptr, b_in, h, N, 1);

  for (int i = 0; i < 3; ++i) {
    gemm128_wmma<<<cdiv(N, 128), blk, 0, stream>>>(h, conv_w + (size_t)i * DH * DH,
                                                   nullptr, nullptr, nullptr, m, N, 0);
    scatter_init_kernel<<<cdiv(N * 32, 256), blk, 0, stream>>>(m, dinv, conv_b + i * DH, agg, N);
    scatter_edge_kernel<<<cdiv(E * 32, 256), blk, 0, stream>>>(m, src, dst, dinv, agg, E);
    fill_kernel<<<1, blk, 0, stream>>>(sums, 0.0f, 2 * DH);
    bn_stats_kernel<<<cdiv(N, 256), blk, 0, stream>>>(agg, sums, N);
    bn_apply_kernel<<<cdiv(N * 32, 256), blk, 0, stream>>>(agg, sums, bn_g + i * DH,
                                                           bn_b + i * DH, h, N, i > 0 ? 1 : 0);
  }

  // pooling
  fill_kernel<<<cdiv(G * DH, 256), blk, 0, stream>>>(gsum, 0.0f, G * DH);
  fill_kernel<<<cdiv(G * DH, 256), blk, 0, stream>>>(gmaxb, 0.0f, G * DH);
  fill_kernel<<<cdiv(G, 256), blk, 0, stream>>>(gcnt, 0.0f, G);
  pool_scatter_kernel<<<cdiv(N * 32, 256), blk, 0, stream>>>(h, batch, gsum, gmaxb, gcnt, N);
  pool_final_kernel<<<cdiv(G * DH, 256), blk, 0, stream>>>(gsum, gcnt, gmean, G * DH);

  // fc1: relu(concat(mean,max) @ fc1_w + b) == relu(mean@W[:128] + max@W[128:] + b)
  gemm128_wmma<<<cdiv(G, 128), blk, 0, stream>>>(gmean, fc1_w, gmaxb, fc1_w + DH * DH,
                                                 fc1_b, fc1o, G, 1);
  // fc2
  fc2_kernel<<<cdiv(G * 10, 256), blk, 0, stream>>>(fc1o, fc2_w, fc2_b, (float*)d_out, G);
}